// RocketLeagueGCN_57363583205981
// MI455X (gfx1250) — compile-verified
//
#include <hip/hip_runtime.h>

typedef __attribute__((ext_vector_type(2))) float v2f;
typedef __attribute__((ext_vector_type(8))) float v8f;

namespace {
constexpr int NG   = 262144;  // graphs
constexpr int NPG  = 6;       // nodes per graph
constexpr int EPG  = 30;      // directed edges per graph (fully connected, no self)
constexpr int GPW  = 8;       // graphs per wave
constexpr int WPB  = 4;       // waves per block (wave32)
constexpr int GPB  = GPW * WPB;   // 32 graphs per block
constexpr int NPW  = GPW * NPG;   // 48 nodes per wave
constexpr int WLDS = 3840;        // floats of LDS per wave (aliased regions)
}

__device__ __forceinline__ v8f wmma4(v2f a, v2f b, v8f c) {
  // V_WMMA_F32_16X16X4_F32 : D = A(16x4,f32) x B(4x16,f32) + C(16x16,f32)
  return __builtin_amdgcn_wmma_f32_16x16x4_f32(false, a, false, b, (short)0, c,
                                               false, false);
}

__global__ __launch_bounds__(128)
void rl_gcn_kernel(const float* __restrict__ x,
                   const float* __restrict__ ewg,
                   const float* __restrict__ W1, const float* __restrict__ b1,
                   const float* __restrict__ W2, const float* __restrict__ b2,
                   const float* __restrict__ Wo, const float* __restrict__ bo,
                   const float* __restrict__ Wb, const float* __restrict__ bb,
                   float* __restrict__ out)
{
  __shared__ float lds[WPB * WLDS];            // 61440 B static LDS
  const int lane = threadIdx.x & 31;
  const int wave = threadIdx.x >> 5;
  const int gb   = blockIdx.x * GPB + wave * GPW;  // first graph of this wave

  float* wl   = lds + wave * WLDS;
  float* ew   = wl + 0;     // 240 floats: edge weights (8 graphs x 30)
  float* dnv  = wl + 240;   // 48: deg^-1/2 per node
  float* ahat = wl + 288;   // 288: Ahat[g][d][s] 6x6 per graph
  float* xb   = wl + 0;     // 192: node features (aliases ew, dead by then)
  float* agg1 = wl + 576;   // 192: Ahat @ X   (48 x 4)
  float* h1   = wl + 768;   // 1536: H1 (48 x 32)
  float* m2   = wl + 2304;  // 1536: Ahat @ H1, overwritten in-place by H2
  float* pool = wl + 0;     // 256: pooled (aliases xb, dead by then)

  const int col  = lane & 15;   // N index (and M index for A fragments)
  const int hi   = lane >> 4;
  const int koff = hi * 2;      // K base for A/B fragments (ISA 16x4 f32 layout)

  // ---- persistent per-lane B fragments + bias splats (weights never re-read)
  v2f l1B[2]; v2f l2B[2][8]; float c1[2], c2[2];
#pragma unroll
  for (int nt = 0; nt < 2; ++nt) {
    l1B[nt].x = W1[(koff + 0) * 32 + nt * 16 + col];
    l1B[nt].y = W1[(koff + 1) * 32 + nt * 16 + col];
    c1[nt] = b1[nt * 16 + col];
    c2[nt] = b2[nt * 16 + col];
#pragma unroll
    for (int kc = 0; kc < 8; ++kc) {
      l2B[nt][kc].x = W2[(kc * 4 + koff + 0) * 32 + nt * 16 + col];
      l2B[nt][kc].y = W2[(kc * 4 + koff + 1) * 32 + nt * 16 + col];
    }
  }

  // ---- phase 1: stage edge weights (topology is implicit; edge_index unused)
  for (int i = lane; i < GPW * EPG; i += 32) ew[i] = ewg[gb * EPG + i];
  __syncthreads();

  // ---- phase 2: deg (incl. self-loop) -> rsqrt
  for (int n = lane; n < NPW; n += 32) {
    int g = n / NPG, d = n - g * NPG;
    float deg = 1.0f;
    int eb = g * EPG;
#pragma unroll
    for (int s = 0; s < NPG; ++s) {
      if (s == d) continue;
      deg += ew[eb + s * 5 + (d > s ? d - 1 : d)];
    }
    dnv[n] = rsqrtf(deg);
  }
  __syncthreads();

  // ---- phase 3: Ahat[g][d][s] = dinv[d] * w(s->d) * dinv[s]; diag = dinv^2
  for (int idx = lane; idx < GPW * 36; idx += 32) {
    int g = idx / 36, r = idx - g * 36;
    int d = r / 6, s = r - d * 6;
    float v;
    if (s == d) { float di = dnv[g * 6 + d]; v = di * di; }
    else {
      int e = s * 5 + (d > s ? d - 1 : d);
      v = dnv[g * 6 + d] * ew[g * EPG + e] * dnv[g * 6 + s];
    }
    ahat[idx] = v;
  }
  __syncthreads();

  // ---- phase 4: stage node features (overwrites dead ew region)
  for (int i = lane; i < NPW * 4; i += 32) xb[i] = x[gb * 24 + i];
  __syncthreads();

  // ---- phase 5: agg1 = Ahat @ X (48x4), aggregate-before-GEMM (fewer FLOPs)
  for (int idx = lane; idx < NPW * 4; idx += 32) {
    int row = idx >> 2, c = idx & 3;
    int g = row / 6, d = row - g * 6;
    float s_ = 0.0f;
#pragma unroll
    for (int s = 0; s < 6; ++s)
      s_ += ahat[g * 36 + d * 6 + s] * xb[(g * 6 + s) * 4 + c];
    agg1[idx] = s_;
  }
  __syncthreads();

  // ---- phase 6: H1 = relu(agg1 @ W1 + b1) — 3 M-tiles x 2 N-tiles, K=4 exact
#pragma unroll
  for (int mt = 0; mt < 3; ++mt) {
    v2f a;
    a.x = agg1[(mt * 16 + col) * 4 + koff + 0];
    a.y = agg1[(mt * 16 + col) * 4 + koff + 1];
#pragma unroll
    for (int nt = 0; nt < 2; ++nt) {
      v8f acc;
#pragma unroll
      for (int i = 0; i < 8; ++i) acc[i] = c1[nt];
      acc = wmma4(a, l1B[nt], acc);
#pragma unroll
      for (int r = 0; r < 8; ++r)
        h1[(mt * 16 + hi * 8 + r) * 32 + nt * 16 + col] = fmaxf(acc[r], 0.0f);
    }
  }
  __syncthreads();

  // ---- phase 7: M2 = Ahat @ H1 (block-diagonal 6x6 apply; VALU, LDS broadcast)
  for (int idx = lane; idx < NPW * 32; idx += 32) {
    int row = idx >> 5, c = idx & 31;
    int g = row / 6, d = row - g * 6;
    float s_ = 0.0f;
#pragma unroll
    for (int s = 0; s < 6; ++s)
      s_ += ahat[g * 36 + d * 6 + s] * h1[(g * 6 + s) * 32 + c];
    m2[idx] = s_;
  }
  __syncthreads();

  // ---- phase 8: H2 = relu(M2 @ W2 + b2) — K=32 as 8 WMMA k-steps per tile
#pragma unroll
  for (int mt = 0; mt < 3; ++mt) {
    v8f acc0, acc1;
#pragma unroll
    for (int i = 0; i < 8; ++i) { acc0[i] = c2[0]; acc1[i] = c2[1]; }
    const float* arow = m2 + (mt * 16 + col) * 32;
#pragma unroll
    for (int kc = 0; kc < 8; ++kc) {
      v2f a;
      a.x = arow[kc * 4 + koff + 0];
      a.y = arow[kc * 4 + koff + 1];
      acc0 = wmma4(a, l2B[0][kc], acc0);
      acc1 = wmma4(a, l2B[1][kc], acc1);
    }
    // store H2 in place over consumed M2 rows (same wave, DS in-order)
#pragma unroll
    for (int r = 0; r < 8; ++r) {
      int row = mt * 16 + hi * 8 + r;
      m2[row * 32 + col]      = fmaxf(acc0[r], 0.0f);
      m2[row * 32 + 16 + col] = fmaxf(acc1[r], 0.0f);
    }
  }
  __syncthreads();

  // ---- phase 9: global mean pool over the 6 nodes of each graph
  for (int idx = lane; idx < GPW * 32; idx += 32) {
    int g = idx >> 5, c = idx & 31;
    float s_ = 0.0f;
#pragma unroll
    for (int d = 0; d < 6; ++d) s_ += m2[(g * 6 + d) * 32 + c];
    pool[idx] = s_ * (1.0f / 6.0f);
  }
  __syncthreads();

  // ---- phase 10: two sigmoid heads (orange first, then blue, flat-concat)
  if (lane < 16) {
    int g = lane >> 1, which = lane & 1;
    const float* Ws = which ? Wb : Wo;
    float acc = which ? bb[0] : bo[0];
#pragma unroll
    for (int c = 0; c < 32; ++c) acc += pool[g * 32 + c] * Ws[c];
    out[(which ? NG : 0) + gb + g] = 1.0f / (1.0f + __expf(-acc));
  }
}

extern "C" void kernel_launch(void* const* d_in, const int* in_sizes, int n_in,
                              void* d_out, int out_size, void* d_ws, size_t ws_size,
                              hipStream_t stream) {
  (void)in_sizes; (void)n_in; (void)out_size; (void)d_ws; (void)ws_size;
  const float* x  = (const float*)d_in[0];
  // d_in[1] = edge_index, d_in[3] = batch : topology is implicit, never read
  const float* ew = (const float*)d_in[2];
  const float* W1 = (const float*)d_in[4];
  const float* b1 = (const float*)d_in[5];
  const float* W2 = (const float*)d_in[6];
  const float* b2 = (const float*)d_in[7];
  const float* Wo = (const float*)d_in[8];
  const float* bo = (const float*)d_in[9];
  const float* Wb = (const float*)d_in[10];
  const float* bb = (const float*)d_in[11];
  float* out = (float*)d_out;

  dim3 grid(NG / GPB);   // 8192 blocks x 128 threads (4 waves, 32 graphs each)
  rl_gcn_kernel<<<grid, 128, 0, stream>>>(x, ew, W1, b1, W2, b2, Wo, bo, Wb, bb, out);
}